// MambaBlock_82291573391911
// MI455X (gfx1250) — compile-verified
//
#include <hip/hip_runtime.h>
#include <hip/hip_bf16.h>

// Mamba block for MI455X (gfx1250, wave32).
//  - fp32 WMMA (V_WMMA_F32_16X16X4_F32) for the two large GEMMs
//  - GLOBAL_LOAD_ASYNC_TO_LDS_B128 double-buffered tile pipeline (ASYNCcnt)
//  - 128x64 block tile, 32x32 per wave: 1.0 LDS fragment load per WMMA
//  - chunked (blocked) selective scan: 16x more parallelism on the recurrence

typedef float v2f __attribute__((ext_vector_type(2)));
typedef float v8f __attribute__((ext_vector_type(8)));

#define DIMD   1024
#define DSN    16
#define DCK    4
#define DID    2048
#define BATCH  2
#define LSEQ   1024
#define NROWS  (BATCH * LSEQ)   // 2048
#define NCH    16               // scan chunks
#define CLEN   (LSEQ / NCH)     // 64 steps per chunk

#define AS3 __attribute__((address_space(3)))

// LDS byte-offset of a __shared__ object (generic -> AS3 -> int).
__device__ __forceinline__ unsigned lds_off(const void* p) {
    return (unsigned)(size_t)(AS3 const void*)p;
}
// Async DMA: 16 bytes global -> LDS, per-lane addresses, tracked by ASYNCcnt.
__device__ __forceinline__ void async_ld16(unsigned dst_lds, const float* src) {
    asm volatile("global_load_async_to_lds_b128 %0, %1, off"
                 :: "v"(dst_lds), "v"((unsigned long long)(size_t)src)
                 : "memory");
}

// --------------------------------------------------------------------------
// LayerNorm: one block per row of 1024.
// --------------------------------------------------------------------------
__global__ __launch_bounds__(256) void ln_kernel(const float* __restrict__ x,
                                                 const float* __restrict__ w,
                                                 const float* __restrict__ b,
                                                 float* __restrict__ xn) {
    __shared__ float red[256];
    const int row = blockIdx.x;
    const float* xr = x + (size_t)row * DIMD;
    float s = 0.f;
    for (int i = threadIdx.x; i < DIMD; i += 256) s += xr[i];
    red[threadIdx.x] = s; __syncthreads();
    for (int o = 128; o > 0; o >>= 1) {
        if (threadIdx.x < o) red[threadIdx.x] += red[threadIdx.x + o];
        __syncthreads();
    }
    const float mu = red[0] * (1.0f / DIMD);
    __syncthreads();
    float v = 0.f;
    for (int i = threadIdx.x; i < DIMD; i += 256) { float d = xr[i] - mu; v += d * d; }
    red[threadIdx.x] = v; __syncthreads();
    for (int o = 128; o > 0; o >>= 1) {
        if (threadIdx.x < o) red[threadIdx.x] += red[threadIdx.x + o];
        __syncthreads();
    }
    const float rstd = rsqrtf(red[0] * (1.0f / DIMD) + 1e-5f);
    float* xo = xn + (size_t)row * DIMD;
    for (int i = threadIdx.x; i < DIMD; i += 256)
        xo[i] = (xr[i] - mu) * rstd * w[i] + b[i];
}

// --------------------------------------------------------------------------
// C[M,N] = A[M,K] * W[N,K]^T (+ optional residual), fp32 WMMA 16x16x4.
// 256 threads = 8 wave32. Block tile 128x64; wave grid 4(M) x 2(N); each wave
// computes a 32x32 patch (2x2 subtiles, 4 v8f accumulators): 16 WMMAs and
// 16 fragment loads per BK=16 chunk per wave. Tiles are row-major in LDS
// (pad 20), double-buffered, filled by global_load_async_to_lds_b128
// (3 x 16B per thread per tile; issue tile t+1 while computing tile t).
// A frag (16x4 f32): lane = m + 16*(k/2), vgpr = k%2   (ISA 7.12.2)
// B frag (4x16 f32): lane = n + 16*(k/2), vgpr = k%2 -> row read of W[n][k..k+1]
// C/D (16x16 f32):   vgpr r holds M = r + 8*(lane/16), N = lane%16
// --------------------------------------------------------------------------
__global__ __launch_bounds__(256) void gemm_wmma_f32(
        const float* __restrict__ A, const float* __restrict__ W,
        const float* __restrict__ res, float* __restrict__ C,
        int M, int N, int K) {
    __shared__ float As[2][128][20];
    __shared__ float Ws[2][64][20];
    const int n0   = blockIdx.x * 64;
    const int m0   = blockIdx.y * 128;
    const int t    = threadIdx.x;
    const int lane = t & 31;
    const int wv   = t >> 5;
    const int wm   = wv & 3;          // wave row: M subtile pair base 2*wm
    const int wn   = wv >> 2;         // wave col: N subtile pair base 2*wn
    const int lm   = lane & 15;
    const int lh   = lane >> 4;

    v8f acc00 = {}, acc01 = {}, acc10 = {}, acc11 = {};

    // DMA slots: thread t moves A rows r and r+64, and W row r (16B each).
    const int r    = t >> 2;          // 0..63
    const int koff = (t & 3) << 2;    // 0,4,8,12
    const float* gA0 = A + (size_t)(m0 + r) * K + koff;
    const float* gA1 = gA0 + (size_t)64 * K;
    const float* gW  = W + (size_t)(n0 + r) * K + koff;

    const int KT = K >> 4;            // BK = 16

    // prologue: issue tile 0
    async_ld16(lds_off(&As[0][r][koff]),      gA0);
    async_ld16(lds_off(&As[0][r + 64][koff]), gA1);
    async_ld16(lds_off(&Ws[0][r][koff]),      gW);

    for (int it = 0; it < KT; ++it) {
        const int cur = it & 1;
        if (it + 1 < KT) {
            const int kp = (it + 1) << 4;
            async_ld16(lds_off(&As[cur ^ 1][r][koff]),      gA0 + kp);
            async_ld16(lds_off(&As[cur ^ 1][r + 64][koff]), gA1 + kp);
            async_ld16(lds_off(&Ws[cur ^ 1][r][koff]),      gW + kp);
            asm volatile("s_wait_asynccnt 0x3" ::: "memory");  // current tile done
        } else {
            asm volatile("s_wait_asynccnt 0x0" ::: "memory");
        }
        __syncthreads();

#pragma unroll
        for (int kk = 0; kk < 4; ++kk) {
            const int kb = kk * 4 + 2 * lh;
            v2f a0, a1, b0, b1;
            a0.x = As[cur][(2 * wm + 0) * 16 + lm][kb];
            a0.y = As[cur][(2 * wm + 0) * 16 + lm][kb + 1];
            a1.x = As[cur][(2 * wm + 1) * 16 + lm][kb];
            a1.y = As[cur][(2 * wm + 1) * 16 + lm][kb + 1];
            b0.x = Ws[cur][(2 * wn + 0) * 16 + lm][kb];
            b0.y = Ws[cur][(2 * wn + 0) * 16 + lm][kb + 1];
            b1.x = Ws[cur][(2 * wn + 1) * 16 + lm][kb];
            b1.y = Ws[cur][(2 * wn + 1) * 16 + lm][kb + 1];
            acc00 = __builtin_amdgcn_wmma_f32_16x16x4_f32(
                false, a0, false, b0, (short)0, acc00, false, false);
            acc01 = __builtin_amdgcn_wmma_f32_16x16x4_f32(
                false, a0, false, b1, (short)0, acc01, false, false);
            acc10 = __builtin_amdgcn_wmma_f32_16x16x4_f32(
                false, a1, false, b0, (short)0, acc10, false, false);
            acc11 = __builtin_amdgcn_wmma_f32_16x16x4_f32(
                false, a1, false, b1, (short)0, acc11, false, false);
        }
        __syncthreads();   // all waves done reading 'cur' before it is refilled
    }

    const int crow0 = m0 + (2 * wm + 0) * 16 + 8 * lh;
    const int crow1 = m0 + (2 * wm + 1) * 16 + 8 * lh;
    const int c0    = n0 + (2 * wn + 0) * 16 + lm;
    const int c1    = n0 + (2 * wn + 1) * 16 + lm;
#pragma unroll
    for (int rr = 0; rr < 8; ++rr) {
        const size_t i00 = (size_t)(crow0 + rr) * N + c0;
        const size_t i01 = (size_t)(crow0 + rr) * N + c1;
        const size_t i10 = (size_t)(crow1 + rr) * N + c0;
        const size_t i11 = (size_t)(crow1 + rr) * N + c1;
        float v00 = acc00[rr], v01 = acc01[rr];
        float v10 = acc10[rr], v11 = acc11[rr];
        if (res) {
            v00 += res[i00]; v01 += res[i01];
            v10 += res[i10]; v11 += res[i11];
        }
        C[i00] = v00; C[i01] = v01;
        C[i10] = v10; C[i11] = v11;
    }
}

// --------------------------------------------------------------------------
// Depthwise causal conv (K=4) + SiLU over xi = xz[..., :DI].
// --------------------------------------------------------------------------
__global__ __launch_bounds__(256) void conv_silu_kernel(
        const float* __restrict__ xz, const float* __restrict__ cw,
        const float* __restrict__ cb, float* __restrict__ xc) {
    const int row = blockIdx.x;          // b*L + l
    const int l   = row & (LSEQ - 1);
    for (int d = threadIdx.x; d < DID; d += 256) {
        float acc = cb[d];
#pragma unroll
        for (int j = 0; j < DCK; ++j) {
            const int lj = l - (DCK - 1) + j;
            if (lj >= 0)
                acc += xz[(size_t)(row - (DCK - 1) + j) * (2 * DID) + d] * cw[d * DCK + j];
        }
        const float sg = 1.0f / (1.0f + __expf(-acc));
        xc[(size_t)row * DID + d] = acc * sg;
    }
}

// --------------------------------------------------------------------------
// x_proj: xp[row][p] = dot(xc[row,:], x_proj_w[p,:]), p < 33. Skinny GEMV.
// --------------------------------------------------------------------------
__global__ __launch_bounds__(256) void xproj_kernel(
        const float* __restrict__ xc, const float* __restrict__ xpw,
        float* __restrict__ xp) {
    __shared__ float rowbuf[DID];
    __shared__ float red[256];
    const int row = blockIdx.x;
    for (int i = threadIdx.x; i < DID; i += 256)
        rowbuf[i] = xc[(size_t)row * DID + i];
    __syncthreads();
    for (int p = 0; p < 2 * DSN + 1; ++p) {
        const float* wp = xpw + (size_t)p * DID;
        float s = 0.f;
        for (int i = threadIdx.x; i < DID; i += 256) s += rowbuf[i] * wp[i];
        red[threadIdx.x] = s; __syncthreads();
        for (int o = 128; o > 0; o >>= 1) {
            if (threadIdx.x < o) red[threadIdx.x] += red[threadIdx.x + o];
            __syncthreads();
        }
        if (threadIdx.x == 0) xp[(size_t)row * (2 * DSN + 1) + p] = red[0];
        __syncthreads();
    }
}

// --------------------------------------------------------------------------
// Blocked selective scan. Recurrence h_l = dA_l*h_{l-1} + dB_l is diagonal in
// (d,n), so per 64-step chunk: P = prod(dA), Q = local scan from h=0. Then a
// tiny sequential fix-up over the 16 chunks yields each chunk's entry state,
// and the chunk replay (pass 3) produces y with full parallelism.
// --------------------------------------------------------------------------
__device__ __forceinline__ float softplus_f(float x) {
    return (x > 20.f) ? x : __logf(1.0f + __expf(x));
}

__global__ __launch_bounds__(256) void scan_pass1(
        const float* __restrict__ xp, const float* __restrict__ xc,
        const float* __restrict__ dt_w, const float* __restrict__ dt_b,
        const float* __restrict__ A_log, float* __restrict__ P,
        float* __restrict__ Q) {
    const int d = blockIdx.x * 256 + threadIdx.x;
    const int c = blockIdx.y;
    const int b = blockIdx.z;
    float Ac[DSN], p[DSN], h[DSN];
#pragma unroll
    for (int n = 0; n < DSN; ++n) {
        Ac[n] = -__expf(A_log[d * DSN + n]);
        p[n] = 1.f; h[n] = 0.f;
    }
    const float dtw = dt_w[d], dtb = dt_b[d];
    for (int l = c * CLEN; l < (c + 1) * CLEN; ++l) {
        const size_t row = (size_t)b * LSEQ + l;
        const float* xpr = xp + row * (2 * DSN + 1);
        const float delta = softplus_f(xpr[0] * dtw + dtb);
        const float dx = delta * xc[row * DID + d];
#pragma unroll
        for (int n = 0; n < DSN; ++n) {
            const float dA = __expf(delta * Ac[n]);
            p[n] *= dA;
            h[n] = dA * h[n] + dx * xpr[1 + n];
        }
    }
    const size_t base = ((size_t)(b * NCH + c) * DSN) * DID + d;
#pragma unroll
    for (int n = 0; n < DSN; ++n) {
        P[base + (size_t)n * DID] = p[n];
        Q[base + (size_t)n * DID] = h[n];
    }
}

__global__ __launch_bounds__(256) void scan_pass2(
        const float* __restrict__ P, const float* __restrict__ Q,
        float* __restrict__ Hs) {
    const int d = blockIdx.x * 256 + threadIdx.x;
    const int b = blockIdx.z;
    float h[DSN];
#pragma unroll
    for (int n = 0; n < DSN; ++n) h[n] = 0.f;
    for (int c = 0; c < NCH; ++c) {
        const size_t base = ((size_t)(b * NCH + c) * DSN) * DID + d;
#pragma unroll
        for (int n = 0; n < DSN; ++n) {
            const size_t i = base + (size_t)n * DID;
            Hs[i] = h[n];                       // entry state of chunk c
            h[n] = P[i] * h[n] + Q[i];
        }
    }
}

__global__ __launch_bounds__(256) void scan_pass3(
        const float* __restrict__ xp, const float* __restrict__ xc,
        const float* __restrict__ xz, const float* __restrict__ dt_w,
        const float* __restrict__ dt_b, const float* __restrict__ A_log,
        const float* __restrict__ D_skip, const float* __restrict__ Hs,
        float* __restrict__ ymod) {
    const int d = blockIdx.x * 256 + threadIdx.x;
    const int c = blockIdx.y;
    const int b = blockIdx.z;
    float Ac[DSN], h[DSN];
    const size_t base = ((size_t)(b * NCH + c) * DSN) * DID + d;
#pragma unroll
    for (int n = 0; n < DSN; ++n) {
        Ac[n] = -__expf(A_log[d * DSN + n]);
        h[n]  = Hs[base + (size_t)n * DID];
    }
    const float dtw = dt_w[d], dtb = dt_b[d], dsk = D_skip[d];
    for (int l = c * CLEN; l < (c + 1) * CLEN; ++l) {
        const size_t row = (size_t)b * LSEQ + l;
        const float* xpr = xp + row * (2 * DSN + 1);
        const float delta = softplus_f(xpr[0] * dtw + dtb);
        const float xv = xc[row * DID + d];
        const float dx = delta * xv;
        float y = 0.f;
#pragma unroll
        for (int n = 0; n < DSN; ++n) {
            const float dA = __expf(delta * Ac[n]);
            h[n] = dA * h[n] + dx * xpr[1 + n];
            y += h[n] * xpr[1 + DSN + n];
        }
        const float z  = xz[row * (2 * DID) + DID + d];
        const float sg = 1.0f / (1.0f + __expf(-z));
        ymod[row * DID + d] = (y + xv * dsk) * (z * sg);
    }
}

// --------------------------------------------------------------------------
extern "C" void kernel_launch(void* const* d_in, const int* in_sizes, int n_in,
                              void* d_out, int out_size, void* d_ws, size_t ws_size,
                              hipStream_t stream) {
    const float* x          = (const float*)d_in[0];
    const float* norm_w     = (const float*)d_in[1];
    const float* norm_b     = (const float*)d_in[2];
    const float* in_proj_w  = (const float*)d_in[3];   // (2*DI, DIM)
    const float* conv_w     = (const float*)d_in[4];   // (DI, 1, DC)
    const float* conv_b     = (const float*)d_in[5];
    const float* x_proj_w   = (const float*)d_in[6];   // (33, DI)
    const float* dt_w       = (const float*)d_in[7];   // (DI, 1)
    const float* dt_b       = (const float*)d_in[8];
    const float* A_log      = (const float*)d_in[9];   // (DI, DS)
    const float* D_skip     = (const float*)d_in[10];
    const float* out_proj_w = (const float*)d_in[11];  // (DIM, DI)
    float* out = (float*)d_out;

    // Workspace layout (floats), ~84MB total — L2-resident on MI455X (192MB).
    float* ws = (float*)d_ws;
    float* xn = ws;                                   // 2048*1024
    float* xz = xn + (size_t)NROWS * DIMD;            // 2048*4096
    float* xc = xz + (size_t)NROWS * 2 * DID;         // 2048*2048
    float* xp = xc + (size_t)NROWS * DID;             // 2048*33
    float* ym = xp + (size_t)NROWS * (2 * DSN + 1);   // 2048*2048
    float* P  = ym + (size_t)NROWS * DID;             // B*NCH*DS*DI
    float* Q  = P  + (size_t)BATCH * NCH * DSN * DID;
    float* Hs = Q  + (size_t)BATCH * NCH * DSN * DID;

    // 1) LayerNorm
    ln_kernel<<<NROWS, 256, 0, stream>>>(x, norm_w, norm_b, xn);
    // 2) in_proj GEMM (M=2048, N=4096, K=1024) — fp32 WMMA + async LDS DMA
    gemm_wmma_f32<<<dim3((2 * DID) / 64, NROWS / 128), 256, 0, stream>>>(
        xn, in_proj_w, nullptr, xz, NROWS, 2 * DID, DIMD);
    // 3) depthwise conv + SiLU
    conv_silu_kernel<<<NROWS, 256, 0, stream>>>(xz, conv_w, conv_b, xc);
    // 4) x_proj skinny GEMV (N=33)
    xproj_kernel<<<NROWS, 256, 0, stream>>>(xc, x_proj_w, xp);
    // 5) blocked selective scan (+ D-skip, z-gate)
    scan_pass1<<<dim3(DID / 256, NCH, BATCH), 256, 0, stream>>>(
        xp, xc, dt_w, dt_b, A_log, P, Q);
    scan_pass2<<<dim3(DID / 256, 1, BATCH), 256, 0, stream>>>(P, Q, Hs);
    scan_pass3<<<dim3(DID / 256, NCH, BATCH), 256, 0, stream>>>(
        xp, xc, xz, dt_w, dt_b, A_log, D_skip, Hs, ym);
    // 6) out_proj GEMM (M=2048, N=1024, K=2048) + residual x — fp32 WMMA
    gemm_wmma_f32<<<dim3(DIMD / 64, NROWS / 128), 256, 0, stream>>>(
        ym, out_proj_w, x, out, NROWS, DIMD, DID);
}